// RGCN_DGL_16449724744364
// MI455X (gfx1250) — compile-verified
//
#include <hip/hip_runtime.h>

// ---- problem constants (match reference) ----
#define N_NODES 100000
#define NREL    8
#define DIM_IN  64
#define DIM_HID 64
#define DIM_OUT 32
#define TILES_M (N_NODES / 16)   // 6250, exact

typedef float v2f __attribute__((ext_vector_type(2)));
typedef float v8f __attribute__((ext_vector_type(8)));

// ============================================================================
// WMMA GEMM: for rr in [0, NREL]:  out = X @ B   (B = W[rr] or loopW if rr==NREL)
//   X: [N_NODES, K] row-major;  W: [NREL, K, H];  T: [NREL, N_NODES, H];
//   selfOut: [N_NODES, H]
// One wave computes a 16(M) x H strip using V_WMMA_F32_16X16X4_F32.
// Layouts per CDNA5 ISA 7.12.2 (32-bit A 16x4, 32-bit B 4x16, 32-bit C/D 16x16).
// ============================================================================
template <int K, int H>
__global__ __launch_bounds__(256)
void rgcn_gemm_wmma(const float* __restrict__ X,
                    const float* __restrict__ W,
                    const float* __restrict__ loopW,
                    float* __restrict__ T,
                    float* __restrict__ selfOut)
{
    const int lane = threadIdx.x & 31;
    const int waveInBlk = threadIdx.x >> 5;
    const int gw = blockIdx.x * (blockDim.x >> 5) + waveInBlk;
    const int totalWaves = TILES_M * (NREL + 1);
    if (gw >= totalWaves) return;               // wave-uniform: EXEC stays all-ones

    const int tileM = gw % TILES_M;
    const int rr    = gw / TILES_M;

    const float* B = (rr < NREL) ? (W + (size_t)rr * K * H) : loopW;
    float* outBase = (rr < NREL)
                   ? (T + ((size_t)rr * N_NODES + (size_t)tileM * 16) * H)
                   : (selfOut + (size_t)tileM * 16 * H);

    const int m     = lane & 15;     // A row / (also B col id)
    const int khalf = lane >> 4;     // 0: K pair {k,k+1}; 1: {k+2,k+3}
    const int n     = lane & 15;     // B/D column within 16-tile

    const float* Arow = X + (size_t)(tileM * 16 + m) * K;

    v8f acc[H / 16];
    const v8f vzero = {0.f, 0.f, 0.f, 0.f, 0.f, 0.f, 0.f, 0.f};
#pragma unroll
    for (int t = 0; t < H / 16; ++t) acc[t] = vzero;

#pragma unroll
    for (int k0 = 0; k0 < K; k0 += 4) {
        const int ka = k0 + 2 * khalf;
        // A fragment: lane holds A[m][ka], A[m][ka+1]  (8B-aligned pair)
        v2f a = *(const v2f*)(Arow + ka);
#pragma unroll
        for (int t = 0; t < H / 16; ++t) {
            v2f b;
            b.x = B[(size_t)ka * H + t * 16 + n];
            b.y = B[(size_t)(ka + 1) * H + t * 16 + n];
            acc[t] = __builtin_amdgcn_wmma_f32_16x16x4_f32(
                /*neg_a=*/false, a, /*neg_b=*/false, b,
                /*c_mod=*/(short)0, acc[t],
                /*reuse_a=*/false, /*reuse_b=*/false);
        }
    }

    // D 16x16 f32 layout: row = j + 8*khalf, col = n
#pragma unroll
    for (int t = 0; t < H / 16; ++t) {
#pragma unroll
        for (int j = 0; j < 8; ++j) {
            outBase[(size_t)(j + 8 * khalf) * H + t * 16 + n] = acc[t][j];
        }
    }
}

// ============================================================================
// Edge gather/scale/scatter:  agg[dst] += t[etype][src] * norm
// One thread per (edge, 4-float chunk): float4 gather + 4x atomicAdd f32.
// ============================================================================
template <int H, int LOG2C>   // H/4 chunks per edge; LOG2C = log2(H/4)
__global__ __launch_bounds__(256)
void rgcn_edges(const float* __restrict__ T,
                const int* __restrict__ etypes,
                const int* __restrict__ src,
                const int* __restrict__ dst,
                const float* __restrict__ norm,
                float* __restrict__ agg,
                int E)
{
    const long tid = (long)blockIdx.x * blockDim.x + threadIdx.x;
    const long e   = tid >> LOG2C;
    const int  c   = (int)(tid & ((H / 4) - 1));
    if (e >= E) return;

    const int   r = etypes[e];
    const int   s = src[e];
    const int   d = dst[e];
    const float w = norm[e];

    const float4 v = *(const float4*)(T + ((size_t)r * N_NODES + (size_t)s) * H + c * 4);
    float* o = agg + (size_t)d * H + c * 4;
    atomicAdd(o + 0, v.x * w);
    atomicAdd(o + 1, v.y * w);
    atomicAdd(o + 2, v.z * w);
    atomicAdd(o + 3, v.w * w);
}

// ============================================================================
// Combine: layer-1 x = relu(agg + self + bias)  (in-place into agg)
// ============================================================================
template <int H>
__global__ __launch_bounds__(256)
void rgcn_combine_relu(const float* __restrict__ selfB,
                       const float* __restrict__ bias,
                       float* __restrict__ agg,
                       long total)
{
    const long i = (long)blockIdx.x * blockDim.x + threadIdx.x;
    if (i >= total) return;
    float v = agg[i] + selfB[i] + bias[(int)(i & (H - 1))];
    agg[i] = v > 0.f ? v : 0.f;
}

// Combine: layer-2 out = agg + self + bias
template <int H>
__global__ __launch_bounds__(256)
void rgcn_combine_out(const float* __restrict__ agg,
                      const float* __restrict__ selfB,
                      const float* __restrict__ bias,
                      float* __restrict__ out,
                      long total)
{
    const long i = (long)blockIdx.x * blockDim.x + threadIdx.x;
    if (i >= total) return;
    out[i] = agg[i] + selfB[i] + bias[(int)(i & (H - 1))];
}

// ============================================================================
// Host side
// ============================================================================
extern "C" void kernel_launch(void* const* d_in, const int* in_sizes, int n_in,
                              void* d_out, int out_size, void* d_ws, size_t ws_size,
                              hipStream_t stream)
{
    (void)n_in; (void)out_size; (void)ws_size;

    const float* features = (const float*)d_in[0];   // [N, 64]
    const int*   etypes   = (const int*)d_in[1];     // [E]
    const float* norm     = (const float*)d_in[2];   // [E]
    const int*   src      = (const int*)d_in[3];     // [E]
    const int*   dst      = (const int*)d_in[4];     // [E]
    const float* W1       = (const float*)d_in[5];   // [8, 64, 64]
    const float* loop1    = (const float*)d_in[6];   // [64, 64]
    const float* b1       = (const float*)d_in[7];   // [64]
    const float* W2       = (const float*)d_in[8];   // [8, 64, 32]
    const float* loop2    = (const float*)d_in[9];   // [64, 32]
    const float* b2       = (const float*)d_in[10];  // [32]
    const int E = in_sizes[1];

    // Workspace carve-up (~269 MB total); t-buffer reused for both layers.
    float* ws      = (float*)d_ws;
    float* t_buf   = ws;                                           // NREL*N*64 (layer1) / NREL*N*32 (layer2)
    float* self_b  = t_buf  + (size_t)NREL * N_NODES * DIM_HID;    // N*64 (layer1) / N*32 (layer2)
    float* agg1    = self_b + (size_t)N_NODES * DIM_HID;           // N*64  -> becomes x after combine
    float* agg2    = agg1   + (size_t)N_NODES * DIM_HID;           // N*32

    const int wavesPerBlk = 256 / 32;
    const int totalWaves  = TILES_M * (NREL + 1);
    const int gemmBlocks  = (totalWaves + wavesPerBlk - 1) / wavesPerBlk;

    // ---------------- Layer 1 ----------------
    hipMemsetAsync(agg1, 0, (size_t)N_NODES * DIM_HID * sizeof(float), stream);

    rgcn_gemm_wmma<DIM_IN, DIM_HID><<<gemmBlocks, 256, 0, stream>>>(
        features, W1, loop1, t_buf, self_b);

    {
        const long threads = (long)E * (DIM_HID / 4);
        const int  blocks  = (int)((threads + 255) / 256);
        rgcn_edges<DIM_HID, 4><<<blocks, 256, 0, stream>>>(
            t_buf, etypes, src, dst, norm, agg1, E);
    }

    {
        const long total = (long)N_NODES * DIM_HID;
        rgcn_combine_relu<DIM_HID><<<(int)((total + 255) / 256), 256, 0, stream>>>(
            self_b, b1, agg1, total);     // agg1 now holds x = relu(...)
    }

    // ---------------- Layer 2 ----------------
    hipMemsetAsync(agg2, 0, (size_t)N_NODES * DIM_OUT * sizeof(float), stream);

    rgcn_gemm_wmma<DIM_HID, DIM_OUT><<<gemmBlocks, 256, 0, stream>>>(
        agg1, W2, loop2, t_buf, self_b);

    {
        const long threads = (long)E * (DIM_OUT / 4);
        const int  blocks  = (int)((threads + 255) / 256);
        rgcn_edges<DIM_OUT, 3><<<blocks, 256, 0, stream>>>(
            t_buf, etypes, src, dst, norm, agg2, E);
    }

    {
        const long total = (long)N_NODES * DIM_OUT;
        rgcn_combine_out<DIM_OUT><<<(int)((total + 255) / 256), 256, 0, stream>>>(
            agg2, self_b, b2, (float*)d_out, total);
    }
}